// GATordered_shared_LSTM_88175678587731
// MI455X (gfx1250) — compile-verified
//
#include <hip/hip_runtime.h>
#include <hip/hip_bf16.h>

// ---------------------------------------------------------------------------
// GOAT (ordered-attention GNN) forward for MI455X / gfx1250, wave32 + WMMA.
// N=512, NFEAT=512, NHID=64, OUTD2=64, NHEADS=4, NCLASS=10.
//
// Pipeline:
//  1) Wh = x @ W_head                       (WMMA f32 16x16x4, K=512)
//  2) s1 = Wh@a[:64], s2 = Wh@a[64:]        (tiny VALU)
//  3) softmax(masked leaky_relu(s1+s2)) + descending bitonic argsort per row
//  4) WhWi = Wh @ gru_Wi[dir]               (WMMA) -- collapses gx GEMM:
//       gx_t[n] = att_s[n,t]*WhWi[idx[n,t]] + bi   (gather, L2-resident)
//  5) GRU scan: async-DMA the step-t gather rows into LDS (ASYNCcnt path)
//     overlapped with gh = h @ gru_Wh via WMMA from LDS; elementwise gates.
//  6) head combine: elu(agg @ lin_W + lin_b) fused into WMMA epilogue
//  7) layer-2 ordered attention (same kernels), final linear + log_softmax
// ---------------------------------------------------------------------------

typedef float v2f __attribute__((ext_vector_type(2)));
typedef float v8f __attribute__((ext_vector_type(8)));
typedef int   b128v __attribute__((vector_size(16)));   // matches async-LDS builtin pointee
typedef __attribute__((address_space(1))) b128v* gp128_t;
typedef __attribute__((address_space(3))) b128v* lp128_t;

#define NN   512
#define HID  64
#define G3   192   // 3*HID gate width
#define NEGV -9.0e15f

#if defined(__has_builtin)
#if __has_builtin(__builtin_amdgcn_global_load_async_to_lds_b128)
#define HAVE_ASYNC_LDS 1
#endif
#endif

// ---------------------------------------------------------------------------
// Generic fp32 WMMA GEMM: C[M,N] = act(A[M,K] @ B[K,N] + bias)
// block = 256 threads = 8 waves in a 2(M) x 4(N) grid of 16x16 tiles.
// grid = (ceil(N/64), ceil(M/32)). Requires M%32==0, N%16==0, K%4==0.
// act: 0 = none, 1 = ELU.
// ---------------------------------------------------------------------------
__global__ __launch_bounds__(256) void wmma_gemm_f32(
    const float* __restrict__ A, const float* __restrict__ B,
    float* __restrict__ C, int M, int N, int K,
    int lda, int ldb, int ldc,
    const float* __restrict__ bias, int act)
{
  const int lane = threadIdx.x & 31;
  const int wave = threadIdx.x >> 5;
  const int row0 = blockIdx.y * 32 + (wave >> 2) * 16;
  const int col0 = blockIdx.x * 64 + (wave & 3) * 16;
  if (row0 >= M || col0 >= N) return;   // wave-uniform exit (EXEC stays full)

  const int half = lane >> 4;           // lanes 16-31 handle K+2/K+3 (A) etc.
  const int lm   = lane & 15;
  const float* Arow = A + (size_t)(row0 + lm) * lda;
  const float* Bcol = B + col0 + lm;

  v8f c = {0.f, 0.f, 0.f, 0.f, 0.f, 0.f, 0.f, 0.f};
  for (int k0 = 0; k0 < K; k0 += 4) {
    const int ka = k0 + (half << 1);
    v2f a, b;
    a.x = Arow[ka];
    a.y = Arow[ka + 1];
    b.x = Bcol[(size_t)ka * ldb];
    b.y = Bcol[(size_t)(ka + 1) * ldb];
    c = __builtin_amdgcn_wmma_f32_16x16x4_f32(false, a, false, b,
                                              (short)0, c, false, false);
  }

  const int col = col0 + lm;
  const float bv = bias ? bias[col] : 0.0f;
  const int rbase = row0 + (half << 3);
  for (int v = 0; v < 8; ++v) {
    float xv = c[v] + bv;
    if (act == 1) xv = (xv > 0.0f) ? xv : (__expf(xv) - 1.0f);
    C[(size_t)(rbase + v) * ldc + col] = xv;
  }
}

// ---------------------------------------------------------------------------
// s1[i] = Wh[i,:] . a[0:64],  s2[i] = Wh[i,:] . a[64:128]
// ---------------------------------------------------------------------------
__global__ void att_vec_kernel(const float* __restrict__ Wh,
                               const float* __restrict__ a,
                               float* __restrict__ s1, float* __restrict__ s2)
{
  int row = blockIdx.x * blockDim.x + threadIdx.x;
  if (row >= NN) return;
  const float* w = Wh + (size_t)row * HID;
  float acc1 = 0.f, acc2 = 0.f;
  for (int d = 0; d < HID; ++d) {
    acc1 += w[d] * a[d];
    acc2 += w[d] * a[HID + d];
  }
  s1[row] = acc1;
  s2[row] = acc2;
}

// ---------------------------------------------------------------------------
// Per row: e = leaky_relu(s1[i]+s2[j], 0.2); masked softmax over j;
// descending bitonic argsort of att. block = 512 threads, grid = 512 rows.
// ---------------------------------------------------------------------------
__global__ __launch_bounds__(512) void att_softsort_kernel(
    const float* __restrict__ s1, const float* __restrict__ s2,
    const float* __restrict__ adj,
    float* __restrict__ att_s, int* __restrict__ idx_out)
{
  __shared__ float key[NN];
  __shared__ int   val[NN];
  __shared__ float red[NN];

  const int row = blockIdx.x;
  const int j   = threadIdx.x;

  float e = s1[row] + s2[j];
  e = (e > 0.f) ? e : 0.2f * e;
  float m = (adj[(size_t)row * NN + j] > 0.f) ? e : NEGV;

  red[j] = m; __syncthreads();
  for (int s = 256; s > 0; s >>= 1) {
    if (j < s) red[j] = fmaxf(red[j], red[j + s]);
    __syncthreads();
  }
  const float mx = red[0]; __syncthreads();

  const float ex = __expf(m - mx);
  red[j] = ex; __syncthreads();
  for (int s = 256; s > 0; s >>= 1) {
    if (j < s) red[j] += red[j + s];
    __syncthreads();
  }
  const float inv = 1.0f / red[0]; __syncthreads();

  key[j] = -(ex * inv);   // ascending sort of -att == descending att
  val[j] = j;
  __syncthreads();

  for (int k = 2; k <= NN; k <<= 1) {
    for (int jj = k >> 1; jj > 0; jj >>= 1) {
      const int ixj = j ^ jj;
      if (ixj > j) {
        const bool up = ((j & k) == 0);
        const float kj = key[j], ki = key[ixj];
        if (up ? (kj > ki) : (kj < ki)) {
          key[j] = ki; key[ixj] = kj;
          int t = val[j]; val[j] = val[ixj]; val[ixj] = t;
        }
      }
      __syncthreads();
    }
  }

  att_s[(size_t)row * NN + j]  = -key[j];
  idx_out[(size_t)row * NN + j] = val[j];
}

// ---------------------------------------------------------------------------
// Bi-GRU scan, one direction. Block handles a 64-node batch tile; h (64x64),
// Wh_gru (64x192) and a gx gather stage (64x192) live in LDS. Per step:
//   (a) issue async DMA (GLOBAL_LOAD_ASYNC_TO_LDS_B128) of the step's
//       att-ordered WhWi rows into the LDS stage,
//   (b) gh = h @ Wh_gru via WMMA f32 (48 16x16 tiles over 8 waves) -- overlaps
//       with the DMA,
//   (c) s_wait_asynccnt 0 + barrier, elementwise GRU gate update.
// dynamic LDS = (64*64 + 3*64*192) * 4 = 163840 bytes (2 blocks / WGP).
// ---------------------------------------------------------------------------
__global__ __launch_bounds__(256) void gru_scan_kernel(
    const float* __restrict__ WhWi,   // [512 x 192] (indexed by source node)
    const float* __restrict__ Whg,    // [64 x 192]
    const float* __restrict__ bi, const float* __restrict__ bh,  // [192]
    const float* __restrict__ att_s, const int* __restrict__ idx, // [512x512]
    float* __restrict__ hout, int ldh, int hoff, int reverse)
{
  extern __shared__ float smem[];
  float* sh_h  = smem;                 // 64*64
  float* sh_gh = smem + 64 * 64;       // 64*192
  float* sh_w  = sh_gh + 64 * G3;      // 64*192
  float* sh_gx = sh_w + 64 * G3;       // 64*192 gather stage

  const int tid   = threadIdx.x;
  const int lane  = tid & 31;
  const int wave  = tid >> 5;
  const int tile0 = blockIdx.x * 64;

  for (int i = tid; i < 64 * G3; i += 256) sh_w[i] = Whg[i];
  for (int i = tid; i < 64 * HID; i += 256) sh_h[i] = 0.0f;
  __syncthreads();

  const int half = lane >> 4;
  const int lm   = lane & 15;

  // elementwise assignment: thread -> (node en, 16 dims starting at ed0)
  const int en  = tid >> 2;
  const int ed0 = (tid & 3) << 4;
  const float* attrow = att_s + (size_t)(tile0 + en) * NN;
  const int*   idxrow = idx   + (size_t)(tile0 + en) * NN;

  for (int t = 0; t < NN; ++t) {
    const int te = reverse ? (NN - 1 - t) : t;

    // ---- (a) stage this step's gx gather rows into LDS ----
#ifdef HAVE_ASYNC_LDS
    for (int c = tid; c < 64 * 48; c += 256) {   // 48 x 16B chunks per row
      const int r  = c / 48;
      const int cc = c - r * 48;
      const int srcn = idx[(size_t)(tile0 + r) * NN + te];
      const float* g = WhWi + (size_t)srcn * G3 + (cc << 2);
      __builtin_amdgcn_global_load_async_to_lds_b128(
          (gp128_t)(uintptr_t)g,
          (lp128_t)(unsigned)(uintptr_t)(sh_gx + r * G3 + (cc << 2)),
          0, 0);
    }
#else
    for (int c = tid * 4; c < 64 * G3; c += 256 * 4) {
      const int r  = c / G3;
      const int cc = c - r * G3;
      const int srcn = idx[(size_t)(tile0 + r) * NN + te];
      const float4 v4 = *(const float4*)(WhWi + (size_t)srcn * G3 + cc);
      *(float4*)(sh_gx + r * G3 + cc) = v4;
    }
    if (t + 1 < NN) {   // prefetch next gather row (global_prefetch_b8)
      const int tnx = reverse ? (NN - 2 - t) : (t + 1);
      __builtin_prefetch(WhWi + (size_t)idxrow[tnx] * G3, 0, 1);
    }
#endif

    // ---- (b) gh = h @ Whg (WMMA f32 from LDS), overlaps DMA ----
    for (int tile = wave; tile < 48; tile += 8) {
      const int tn = tile / 12;            // node sub-tile 0..3
      const int tc = tile % 12;            // gate-col sub-tile 0..11
      const int r0 = tn << 4, c0 = tc << 4;
      const float* hrow = sh_h + (size_t)(r0 + lm) * HID;
      const float* wcol = sh_w + c0 + lm;
      v8f c = {0.f, 0.f, 0.f, 0.f, 0.f, 0.f, 0.f, 0.f};
      for (int k0 = 0; k0 < HID; k0 += 4) {
        const int ka = k0 + (half << 1);
        v2f a, b;
        a.x = hrow[ka];
        a.y = hrow[ka + 1];
        b.x = wcol[ka * G3];
        b.y = wcol[(ka + 1) * G3];
        c = __builtin_amdgcn_wmma_f32_16x16x4_f32(false, a, false, b,
                                                  (short)0, c, false, false);
      }
      const int rr = r0 + (half << 3);
      const int cc = c0 + lm;
      for (int v = 0; v < 8; ++v) sh_gh[(rr + v) * G3 + cc] = c[v];
    }
#ifdef HAVE_ASYNC_LDS
    asm volatile("s_wait_asynccnt 0x0" ::: "memory");
#endif
    __syncthreads();

    // ---- (c) elementwise GRU gates ----
    {
      const float av = attrow[te];
      const float* wrow = sh_gx + en * G3;
      for (int d = ed0; d < ed0 + 16; ++d) {
        const float gxr = av * wrow[d]           + bi[d];
        const float gxz = av * wrow[HID + d]     + bi[HID + d];
        const float gxn = av * wrow[2 * HID + d] + bi[2 * HID + d];
        const float ghr = sh_gh[en * G3 + d]           + bh[d];
        const float ghz = sh_gh[en * G3 + HID + d]     + bh[HID + d];
        const float ghn = sh_gh[en * G3 + 2 * HID + d] + bh[2 * HID + d];
        const float r  = 1.0f / (1.0f + __expf(-(gxr + ghr)));
        const float z  = 1.0f / (1.0f + __expf(-(gxz + ghz)));
        const float nv = tanhf(gxn + r * ghn);
        const float hp = sh_h[en * HID + d];
        sh_h[en * HID + d] = (1.0f - z) * nv + z * hp;
      }
    }
    __syncthreads();
  }

  for (int i = tid; i < 64 * HID; i += 256) {
    const int n = i >> 6, d = i & 63;
    hout[(size_t)(tile0 + n) * ldh + hoff + d] = sh_h[i];
  }
}

// ---------------------------------------------------------------------------
// logits = elu(agg2 @ lin_out_W + b); out = log_softmax(logits). 1 wave/row.
// ---------------------------------------------------------------------------
__global__ __launch_bounds__(32) void final_kernel(
    const float* __restrict__ agg2, const float* __restrict__ W,
    const float* __restrict__ b, float* __restrict__ out)
{
  __shared__ float lg[16];
  const int row = blockIdx.x;
  const int lane = threadIdx.x;
  float v = 0.0f;
  if (lane < 10) {
    float acc = b[lane];
    const float* ar = agg2 + (size_t)row * 128;
    for (int k = 0; k < 128; ++k) acc += ar[k] * W[k * 10 + lane];
    v = (acc > 0.f) ? acc : (__expf(acc) - 1.0f);
    lg[lane] = v;
  }
  __syncthreads();
  float mx = -1e30f;
  for (int k = 0; k < 10; ++k) mx = fmaxf(mx, lg[k]);
  float sum = 0.f;
  for (int k = 0; k < 10; ++k) sum += __expf(lg[k] - mx);
  if (lane < 10) out[row * 10 + lane] = v - mx - __logf(sum);
}

// ---------------------------------------------------------------------------
// Host-side orchestration
// ---------------------------------------------------------------------------
static inline void launch_gemm(const float* A, const float* B, float* C,
                               int M, int N, int K, int lda, int ldb, int ldc,
                               const float* bias, int act, hipStream_t s)
{
  dim3 grid((N + 63) / 64, (M + 31) / 32);
  wmma_gemm_f32<<<grid, 256, 0, s>>>(A, B, C, M, N, K, lda, ldb, ldc, bias, act);
}

extern "C" void kernel_launch(void* const* d_in, const int* in_sizes, int n_in,
                              void* d_out, int out_size, void* d_ws, size_t ws_size,
                              hipStream_t stream)
{
  (void)in_sizes; (void)n_in; (void)out_size; (void)ws_size;

  const float* x      = (const float*)d_in[0];   // 512x512
  const float* adj    = (const float*)d_in[1];   // 512x512
  const float* Wh_w   = (const float*)d_in[2];   // 4x512x64
  const float* a_h    = (const float*)d_in[3];   // 4x128
  const float* linW   = (const float*)d_in[4];   // 4x128x64
  const float* linB   = (const float*)d_in[5];   // 4x64
  const float* g1Wi   = (const float*)d_in[6];   // 2x64x192
  const float* g1Wh   = (const float*)d_in[7];   // 2x64x192
  const float* g1bi   = (const float*)d_in[8];   // 2x192
  const float* g1bh   = (const float*)d_in[9];   // 2x192
  const float* Wout   = (const float*)d_in[10];  // 256x64
  const float* a_out  = (const float*)d_in[11];  // 128
  const float* g2Wi   = (const float*)d_in[12];
  const float* g2Wh   = (const float*)d_in[13];
  const float* g2bi   = (const float*)d_in[14];
  const float* g2bh   = (const float*)d_in[15];
  const float* loW    = (const float*)d_in[16];  // 128x10
  const float* lob    = (const float*)d_in[17];  // 10
  float* out = (float*)d_out;

  // ---- workspace layout (floats) ----
  float* ws = (float*)d_ws;
  size_t off = 0;
  float* Wh1   = ws + off; off += (size_t)4 * NN * HID;        // 4x512x64
  float* s1b   = ws + off; off += (size_t)4 * NN;
  float* s2b   = ws + off; off += (size_t)4 * NN;
  float* att1  = ws + off; off += (size_t)4 * NN * NN;         // 4x512x512
  int*   idx1  = (int*)(ws + off); off += (size_t)4 * NN * NN;
  float* WhWi1 = ws + off; off += (size_t)4 * 2 * NN * G3;     // 4x2x512x192
  float* agg1  = ws + off; off += (size_t)4 * NN * 128;        // 4x512x128
  float* hbuf  = ws + off; off += (size_t)NN * 256;            // 512x256
  float* Wh2   = ws + off; off += (size_t)NN * HID;
  float* s1c   = ws + off; off += NN;
  float* s2c   = ws + off; off += NN;
  float* att2  = ws + off; off += (size_t)NN * NN;
  int*   idx2  = (int*)(ws + off); off += (size_t)NN * NN;
  float* WhWi2 = ws + off; off += (size_t)2 * NN * G3;
  float* agg2  = ws + off; off += (size_t)NN * 128;
  // total ~4.3M floats (~17 MB)

  const size_t SMEM = (size_t)(64 * 64 + 3 * 64 * G3) * sizeof(float); // 160 KB

  // ---------------- layer 1: per-head ordered attention ----------------
  for (int h = 0; h < 4; ++h) {
    float* Wh_h  = Wh1 + (size_t)h * NN * HID;
    float* att_h = att1 + (size_t)h * NN * NN;
    int*   idx_h = idx1 + (size_t)h * NN * NN;
    float* agg_h = agg1 + (size_t)h * NN * 128;

    // Wh = x @ W[h]    (512x512x64)
    launch_gemm(x, Wh_w + (size_t)h * NN * HID, Wh_h,
                NN, HID, NN, NN, HID, HID, nullptr, 0, stream);

    att_vec_kernel<<<4, 128, 0, stream>>>(Wh_h, a_h + h * 128, s1b + h * NN, s2b + h * NN);
    att_softsort_kernel<<<NN, NN, 0, stream>>>(s1b + h * NN, s2b + h * NN, adj, att_h, idx_h);

    for (int dir = 0; dir < 2; ++dir) {
      float* WhWi_hd = WhWi1 + (size_t)(h * 2 + dir) * NN * G3;
      // WhWi = Wh @ gru_Wi[dir]   (512x64x192)
      launch_gemm(Wh_h, g1Wi + (size_t)dir * HID * G3, WhWi_hd,
                  NN, G3, HID, HID, G3, G3, nullptr, 0, stream);
      gru_scan_kernel<<<8, 256, SMEM, stream>>>(
          WhWi_hd, g1Wh + (size_t)dir * HID * G3,
          g1bi + dir * G3, g1bh + dir * G3,
          att_h, idx_h, agg_h, 128, dir * HID, dir);
    }

    // hbuf[:, h*64:(h+1)*64] = elu(agg_h @ linW[h] + linB[h])
    launch_gemm(agg_h, linW + (size_t)h * 128 * HID, hbuf + h * HID,
                NN, HID, 128, 128, HID, 256, linB + h * HID, 1, stream);
  }

  // ---------------- layer 2: ordered attention on concatenated heads ---
  launch_gemm(hbuf, Wout, Wh2, NN, HID, 256, 256, HID, HID, nullptr, 0, stream);
  att_vec_kernel<<<4, 128, 0, stream>>>(Wh2, a_out, s1c, s2c);
  att_softsort_kernel<<<NN, NN, 0, stream>>>(s1c, s2c, adj, att2, idx2);

  for (int dir = 0; dir < 2; ++dir) {
    float* WhWi_d = WhWi2 + (size_t)dir * NN * G3;
    launch_gemm(Wh2, g2Wi + (size_t)dir * HID * G3, WhWi_d,
                NN, G3, HID, HID, G3, G3, nullptr, 0, stream);
    gru_scan_kernel<<<8, 256, SMEM, stream>>>(
        WhWi_d, g2Wh + (size_t)dir * HID * G3,
        g2bi + dir * G3, g2bh + dir * G3,
        att2, idx2, agg2, 128, dir * HID, dir);
  }

  final_kernel<<<NN, 32, 0, stream>>>(agg2, loW, lob, out);
}